// DisSimilarity_Loss_46334107189856
// MI455X (gfx1250) — compile-verified
//
#include <hip/hip_runtime.h>
#include <hip/hip_bf16.h>
#include <math.h>

// Problem constants from the reference: P=20, N=1024, D=512.
#define PP 20
#define NN 1024
#define DD 512
#define EPS 1e-8f
#define COUNT_F 20951040.0f   // P*N*(N-1)

typedef float v2f __attribute__((ext_vector_type(2)));
typedef float v8f __attribute__((ext_vector_type(8)));

// ---------------- Kernel 0: zero the accumulators ----------------
__global__ void dsim_init_kernel(float* __restrict__ Avec, float* __restrict__ acc) {
    int t = threadIdx.x;
    if (t < DD) Avec[t] = 0.0f;
    if (t == 0) acc[0] = 0.0f;
}

// ---------------- Kernel 1: z_avg[n,d] = mean_p z[p,n,d] ----------------
__global__ __launch_bounds__(256) void dsim_mean_kernel(const float* __restrict__ Z,
                                                        float* __restrict__ Zavg) {
    int i = blockIdx.x * 256 + threadIdx.x;     // i in [0, N*D)
    float s = 0.0f;
#pragma unroll
    for (int p = 0; p < PP; ++p)
        s += Z[(size_t)p * (NN * DD) + i];
    Zavg[i] = s * (1.0f / (float)PP);
}

// ------- Kernel 2: per-row rinv of z_avg; A = sum_j z_avg[j]*rinv[j] -------
__global__ __launch_bounds__(256) void dsim_norm_kernel(const float* __restrict__ Zavg,
                                                        float* __restrict__ rinvA,
                                                        float* __restrict__ Avec) {
    __shared__ float red[256];
    __shared__ float srinv;
    int j = blockIdx.x;
    int t = threadIdx.x;
    const float* row = Zavg + (size_t)j * DD;
    float a0 = row[t];
    float a1 = row[t + 256];
    red[t] = a0 * a0 + a1 * a1;
    __syncthreads();
#pragma unroll
    for (int s = 128; s > 0; s >>= 1) {
        if (t < s) red[t] += red[t + s];
        __syncthreads();
    }
    if (t == 0) {
        float rv = 1.0f / fmaxf(sqrtf(red[0]), EPS);
        rinvA[j] = rv;
        srinv = rv;
    }
    __syncthreads();
    float rv = srinv;
    atomicAdd(&Avec[t],       a0 * rv);
    atomicAdd(&Avec[t + 256], a1 * rv);
}

// ---------------- Kernel 3: WMMA tile kernel ----------------
// One wave32 per 16-row tile of z_list (rows share the same p; j0..j0+15).
// Accumulates over K=512 in LDS-staged chunks of 128:
//   cN = Z Z^T      -> diag = ||z_m||^2
//   cD = Z An^T     -> diag = z_m . an_m
//   cA = Z A_bcast  -> any col = z_m . A
// contribution_m = (z.A - z.an_m) / max(||z||, eps); atomicAdd wave sum.
#define TK 128
#define LSTR 132   // 128 + 4 pad: bank = (4*row + k) % 64 -> conflict-free b64 reads

__global__ __launch_bounds__(64) void dsim_tile_kernel(const float* __restrict__ Z,
                                                       const float* __restrict__ Zavg,
                                                       const float* __restrict__ rinvA,
                                                       const float* __restrict__ Avec,
                                                       float* __restrict__ acc) {
    __shared__ float ldsA[DD];
    __shared__ float ldsZ[2][16 * LSTR];
    __shared__ float ldsN[2][16 * LSTR];

    int tid = threadIdx.x;
    for (int i = tid; i < DD; i += 64) ldsA[i] = Avec[i];
    __syncthreads();

    const int wave = tid >> 5;
    const int lane = tid & 31;
    const int tile = blockIdx.x * 2 + wave;      // 1280 tiles total
    const int row0 = tile * 16;                  // global row in [0, P*N)
    const int j0   = row0 & (NN - 1);            // row within N (16 | N)

    const float* zrow = Z    + (size_t)row0 * DD;
    const float* arow = Zavg + (size_t)j0   * DD;
    float* lz = ldsZ[wave];
    float* ln = ldsN[wave];

    v8f cN = {}; v8f cD = {}; v8f cA = {};

    const int rsel  = lane & 15;
    const int khalf = 2 * (lane >> 4);

#pragma unroll 1
    for (int kc = 0; kc < DD; kc += TK) {
        // Stage 16x128 chunks of Z and An (= z_avg * rinv) into LDS, coalesced.
#pragma unroll
        for (int r = 0; r < 16; ++r) {
            float4 vz = *(const float4*)(zrow + (size_t)r * DD + kc + lane * 4);
            *(float4*)(lz + r * LSTR + lane * 4) = vz;
            float rs = rinvA[j0 + r];
            float4 va = *(const float4*)(arow + (size_t)r * DD + kc + lane * 4);
            va.x *= rs; va.y *= rs; va.z *= rs; va.w *= rs;
            *(float4*)(ln + r * LSTR + lane * 4) = va;
        }
        // 32 k-steps of V_WMMA_F32_16X16X4_F32 per chunk, 3 accumulators.
        // A-operand layout (16x4 f32): lane l holds Z[l%16][k0+2*(l/16) .. +1].
        // B-operand of X^T has the identical per-lane pattern -> same loads.
#pragma unroll
        for (int k0 = 0; k0 < TK; k0 += 4) {
            int koff = k0 + khalf;
            v2f a  = *(const v2f*)(lz + rsel * LSTR + koff);
            v2f bD = *(const v2f*)(ln + rsel * LSTR + koff);
            v2f bA = *(const v2f*)(&ldsA[kc + koff]);   // broadcast per half-wave
            cN = __builtin_amdgcn_wmma_f32_16x16x4_f32(false, a, false, a,
                                                       (short)0, cN, false, false);
            cD = __builtin_amdgcn_wmma_f32_16x16x4_f32(false, a, false, bD,
                                                       (short)0, cD, false, false);
            cA = __builtin_amdgcn_wmma_f32_16x16x4_f32(false, a, false, bA,
                                                       (short)0, cA, false, false);
        }
    }

    // Diagonal extraction from the 16x16 f32 C layout:
    //   (m,m): m<8 -> lane m, vgpr m;  m>=8 -> lane m+16, vgpr m-8.
    int idx = -1;
    if (lane < 8)       idx = lane;
    else if (lane >= 24) idx = lane - 24;

    float dN = 0.0f, dD = 0.0f, dA = 0.0f;
    if (idx >= 0) {
#pragma unroll
        for (int i = 0; i < 8; ++i) {
            if (i == idx) { dN = cN[i]; dD = cD[i]; dA = cA[i]; }
        }
    }
    float contrib = 0.0f;
    if (idx >= 0)
        contrib = (dA - dD) / fmaxf(sqrtf(dN), EPS);

    // wave32 reduction
#pragma unroll
    for (int off = 16; off > 0; off >>= 1)
        contrib += __shfl_xor(contrib, off, 32);
    if (lane == 0)
        atomicAdd(acc, contrib);
}

// ---------------- Kernel 4: finalize ----------------
__global__ void dsim_fin_kernel(const float* __restrict__ acc, float* __restrict__ out) {
    out[0] = acc[0] / COUNT_F - 1.0f;
}

// ---------------- launch ----------------
extern "C" void kernel_launch(void* const* d_in, const int* in_sizes, int n_in,
                              void* d_out, int out_size, void* d_ws, size_t ws_size,
                              hipStream_t stream) {
    const float* Z = (const float*)d_in[0];   // [P, N, D] f32; d_in[1] (z_avg) ignored
    float* out = (float*)d_out;

    // ws layout (floats): Zavg[N*D] | rinvA[N] | Avec[D] | acc[1]
    float* ws    = (float*)d_ws;
    float* Zavg  = ws;
    float* rinvA = Zavg + (size_t)NN * DD;
    float* Avec  = rinvA + NN;
    float* acc   = Avec + DD;

    dsim_init_kernel<<<1, 512, 0, stream>>>(Avec, acc);
    dsim_mean_kernel<<<(NN * DD) / 256, 256, 0, stream>>>(Z, Zavg);
    dsim_norm_kernel<<<NN, 256, 0, stream>>>(Zavg, rinvA, Avec);
    dsim_tile_kernel<<<(PP * NN / 16) / 2, 64, 0, stream>>>(Z, Zavg, rinvA, Avec, acc);
    dsim_fin_kernel<<<1, 1, 0, stream>>>(acc, out);
}